// pointcnn_38319698215330
// MI455X (gfx1250) — compile-verified
//
#include <hip/hip_runtime.h>
#include <math.h>

typedef float v2f __attribute__((ext_vector_type(2)));
typedef float v8f __attribute__((ext_vector_type(8)));

#define BSZ   8
#define NPT   2048
#define CH    64
#define KNN   16
#define ROWS  (BSZ*NPT*KNN)   // 262144
#define EPS   1e-5f

// ---------------------------------------------------------------------------
// Kernel 1: KNN.  One block = 64 query points of one batch; whole point cloud
// of that batch staged in LDS (24KB).  Each wave handles 8 queries; per-lane
// top-16 (replace-max), then 16 rounds of cross-lane min extraction.
// ---------------------------------------------------------------------------
__device__ __forceinline__ unsigned long long wave_min_u64(unsigned long long k) {
#pragma unroll
  for (int m = 16; m >= 1; m >>= 1) {
    unsigned lo = (unsigned)k, hi = (unsigned)(k >> 32);
    unsigned olo = __shfl_xor(lo, m, 32);
    unsigned ohi = __shfl_xor(hi, m, 32);
    unsigned long long o = ((unsigned long long)ohi << 32) | olo;
    if (o < k) k = o;
  }
  return k;
}

__global__ __launch_bounds__(256)
void knn_kernel(const float* __restrict__ xyz, int* __restrict__ knn_idx) {
  __shared__ __attribute__((aligned(16))) float sx[NPT];
  __shared__ __attribute__((aligned(16))) float sy[NPT];
  __shared__ __attribute__((aligned(16))) float sz[NPT];
  const int b = blockIdx.x >> 5;          // 32 chunks per batch
  const int chunk = blockIdx.x & 31;
  for (int t = threadIdx.x; t < NPT; t += 256) {
    sx[t] = xyz[(b*3 + 0)*NPT + t];
    sy[t] = xyz[(b*3 + 1)*NPT + t];
    sz[t] = xyz[(b*3 + 2)*NPT + t];
  }
  __syncthreads();
  const int wave = threadIdx.x >> 5, lane = threadIdx.x & 31;
  for (int qi = 0; qi < 8; ++qi) {
    const int q = chunk*64 + wave*8 + qi;
    const float qx = sx[q], qy = sy[q], qz = sz[q];
    float dbest[KNN]; int ibest[KNN];
#pragma unroll
    for (int j = 0; j < KNN; ++j) { dbest[j] = 3.4e38f; ibest[j] = -1; }
    float dmax = 3.4e38f;
    for (int c = lane; c < NPT; c += 32) {
      if (c == q) continue;
      float dx = sx[c]-qx, dy = sy[c]-qy, dz = sz[c]-qz;
      float d2 = fmaf(dx, dx, fmaf(dy, dy, dz*dz));
      if (d2 < dmax) {
        bool done = false;
#pragma unroll
        for (int j = 0; j < KNN; ++j)
          if (!done && dbest[j] == dmax) { dbest[j] = d2; ibest[j] = c; done = true; }
        dmax = dbest[0];
#pragma unroll
        for (int j = 1; j < KNN; ++j) dmax = fmaxf(dmax, dbest[j]);
      }
    }
    // merge: extract the 16 globally smallest (set semantics are enough:
    // downstream ops over k are symmetric)
    for (int r = 0; r < KNN; ++r) {
      float lmin = dbest[0]; int li = ibest[0];
#pragma unroll
      for (int j = 1; j < KNN; ++j)
        if (dbest[j] < lmin) { lmin = dbest[j]; li = ibest[j]; }
      unsigned long long key =
          ((unsigned long long)__float_as_uint(lmin) << 32) | (unsigned)li;
      unsigned long long gk = wave_min_u64(key);
      int gi = (int)(unsigned)gk;
      if (key == gk) {                       // winning lane: retire that slot
        bool done = false;
#pragma unroll
        for (int j = 0; j < KNN; ++j)
          if (!done && ibest[j] == gi) { dbest[j] = 3.4e38f; ibest[j] = -1; done = true; }
      }
      if (lane == 0) knn_idx[((b*NPT + q) << 4) + r] = gi;
    }
  }
}

// ---------------------------------------------------------------------------
// Kernel 2: gather neighbors, relative coords, 3->64 conv (+b1), write h1,
// and per-block BN-stat partials (sum / sumsq per channel).
// ---------------------------------------------------------------------------
__global__ __launch_bounds__(256)
void conv1_kernel(const float* __restrict__ xyz, const int* __restrict__ knn_idx,
                  const float* __restrict__ W1, const float* __restrict__ b1,
                  float* __restrict__ hbuf, float* __restrict__ partial) {
  __shared__ float ssum[2*CH];
  const int tid = threadIdx.x;
  if (tid < 2*CH) ssum[tid] = 0.f;
  __syncthreads();
  const int row = blockIdx.x*256 + tid;         // 262144 rows total
  const int b   = row >> 15;                    // / (N*K)
  const int rem = row & 32767;
  const int n   = rem >> 4;
  const int j   = knn_idx[row];
  const float* xb = xyz + (size_t)b*3*NPT;
  const float rx = xb[j]        - xb[n];
  const float ry = xb[NPT + j]  - xb[NPT + n];
  const float rz = xb[2*NPT+j]  - xb[2*NPT+n];
  float* hrow = hbuf + (size_t)row*CH;
  for (int c0 = 0; c0 < CH; c0 += 4) {
    float4 h;
    float* hp = (float*)&h;
#pragma unroll
    for (int u = 0; u < 4; ++u) {
      const int c = c0 + u;
      hp[u] = fmaf(W1[c*3+2], rz, fmaf(W1[c*3+1], ry, fmaf(W1[c*3+0], rx, b1[c])));
    }
    *(float4*)(hrow + c0) = h;
#pragma unroll
    for (int u = 0; u < 4; ++u) {
      float s = hp[u], qq = hp[u]*hp[u];
#pragma unroll
      for (int m = 16; m >= 1; m >>= 1) {
        s  += __shfl_xor(s,  m, 32);
        qq += __shfl_xor(qq, m, 32);
      }
      if ((tid & 31) == 0) { atomicAdd(&ssum[c0+u], s); atomicAdd(&ssum[CH+c0+u], qq); }
    }
  }
  __syncthreads();
  if (tid < 2*CH) partial[blockIdx.x*2*CH + tid] = ssum[tid];
}

// ---------------------------------------------------------------------------
// Kernel 3: deterministic stat reduction -> per-channel scale/shift
// ---------------------------------------------------------------------------
__global__ void bn_reduce_kernel(const float* __restrict__ partial, int nblocks,
                                 float inv_count, const float* __restrict__ gamma,
                                 const float* __restrict__ beta,
                                 float* __restrict__ stats) {
  const int c = threadIdx.x;                     // 64 threads
  float s = 0.f, q = 0.f;
  for (int i = 0; i < nblocks; ++i) {
    s += partial[i*2*CH + c];
    q += partial[i*2*CH + CH + c];
  }
  const float mean = s * inv_count;
  const float var  = q * inv_count - mean*mean;
  const float sc   = rsqrtf(var + EPS) * gamma[c];
  stats[c]      = sc;
  stats[CH + c] = beta[c] - mean*sc;
}

// ---------------------------------------------------------------------------
// Shared WMMA tile body: 16 rows x 64ch  times  W2^T (64x64) with f32 WMMA.
// A layout (16x4 f32): lane<16 -> {K=k0,k0+1}, lane>=16 -> {K=k0+2,k0+3}.
// B layout (4x16):     VGPR pair mirrors K rows across lane halves; n = lane&15.
// ---------------------------------------------------------------------------
#define WMMA_TILE(acc, atile_row, w2s, lh, hb)                                   \
  _Pragma("unroll")                                                              \
  for (int kc = 0; kc < 16; ++kc) {                                              \
    v2f a = *(const v2f*)&(atile_row)[(lh)*CH + kc*4 + (hb)*2];                  \
    _Pragma("unroll")                                                            \
    for (int nc = 0; nc < 4; ++nc) {                                             \
      v2f w = *(const v2f*)&(w2s)[(nc*16 + (lh))*CH + kc*4 + (hb)*2];            \
      acc[nc] = __builtin_amdgcn_wmma_f32_16x16x4_f32(                           \
          false, a, false, w, (short)0, acc[nc], false, false);                  \
    }                                                                            \
  }

// ---------------------------------------------------------------------------
// Kernel 4: BN1+ReLU -> GEMM(W2)+b2 -> h2 (in place) + BN2 stat partials
// ---------------------------------------------------------------------------
__global__ __launch_bounds__(256)
void conv_mid_kernel(const float* __restrict__ W2g, const float* __restrict__ b2,
                     const float* __restrict__ stats, float* __restrict__ hbuf,
                     float* __restrict__ partial) {
  __shared__ __attribute__((aligned(16))) float w2s[CH*CH];       // 16KB
  __shared__ float sscale[2*CH];
  __shared__ float ssum[2*CH];
  __shared__ __attribute__((aligned(16))) float atile[8][16*CH];  // 32KB
  const int tid = threadIdx.x;
  for (int t = tid; t < CH*CH; t += 256) w2s[t] = W2g[t];
  if (tid < 2*CH) { sscale[tid] = stats[tid]; ssum[tid] = 0.f; }
  __syncthreads();
  const int wave = tid >> 5, lane = tid & 31;
  const int lh = lane & 15, hb = lane >> 4;
  const int rowbase = (blockIdx.x*8 + wave) * 16;
  // stage A tile with fused BN + ReLU
  for (int e = lane; e < 256; e += 32) {
    const int rr = e >> 4, c0 = (e & 15) * 4;
    float4 v = *(const float4*)(hbuf + (size_t)(rowbase + rr)*CH + c0);
    float* vp = (float*)&v;
#pragma unroll
    for (int u = 0; u < 4; ++u)
      vp[u] = fmaxf(fmaf(vp[u], sscale[c0+u], sscale[CH+c0+u]), 0.f);
    *(float4*)&atile[wave][rr*CH + c0] = v;
  }
  __syncthreads();
  v8f acc[4] = {};
  WMMA_TILE(acc, atile[wave], w2s, lh, hb)
  // bias + stats + stage result
#pragma unroll
  for (int nc = 0; nc < 4; ++nc) {
    const int nch = nc*16 + lh;
    const float bias = b2[nch];
    float s = 0.f, q = 0.f;
#pragma unroll
    for (int r = 0; r < 8; ++r) {
      const float val = acc[nc][r] + bias;
      atile[wave][(r + hb*8)*CH + nch] = val;
      s += val; q += val*val;
    }
    s += __shfl_xor(s, 16, 32); q += __shfl_xor(q, 16, 32);
    if (hb == 0) { atomicAdd(&ssum[nch], s); atomicAdd(&ssum[CH+nch], q); }
  }
  __syncthreads();
  for (int e = lane; e < 256; e += 32) {
    const int rr = e >> 4, c0 = (e & 15) * 4;
    *(float4*)(hbuf + (size_t)(rowbase + rr)*CH + c0) =
        *(const float4*)&atile[wave][rr*CH + c0];
  }
  __syncthreads();
  if (tid < 2*CH) partial[blockIdx.x*2*CH + tid] = ssum[tid];
}

// ---------------------------------------------------------------------------
// Kernel 5: BN2+ReLU -> GEMM(W2)+b2 -> max over k -> out [B, C, N]
// One 16-row tile == one point's 16 neighbors, so max-over-k = max over tile rows.
// ---------------------------------------------------------------------------
__global__ __launch_bounds__(256)
void conv_out_kernel(const float* __restrict__ W2g, const float* __restrict__ b2,
                     const float* __restrict__ stats, const float* __restrict__ hbuf,
                     float* __restrict__ out) {
  __shared__ __attribute__((aligned(16))) float w2s[CH*CH];
  __shared__ float sscale[2*CH];
  __shared__ __attribute__((aligned(16))) float res[16*CH];
  __shared__ __attribute__((aligned(16))) float atile[8][16*CH];
  const int tid = threadIdx.x;
  for (int t = tid; t < CH*CH; t += 256) w2s[t] = W2g[t];
  if (tid < 2*CH) sscale[tid] = stats[tid];
  __syncthreads();
  const int wave = tid >> 5, lane = tid & 31;
  const int lh = lane & 15, hb = lane >> 4;
  const int g0 = blockIdx.x * 16;              // 16 consecutive points per block
  const int b  = g0 >> 11;
  const int nbase = g0 & 2047;
  for (int gi = 0; gi < 2; ++gi) {
    const int lp = wave*2 + gi;                // local point 0..15
    const int rowbase = (g0 + lp) * 16;
    for (int e = lane; e < 256; e += 32) {
      const int rr = e >> 4, c0 = (e & 15) * 4;
      float4 v = *(const float4*)(hbuf + (size_t)(rowbase + rr)*CH + c0);
      float* vp = (float*)&v;
#pragma unroll
      for (int u = 0; u < 4; ++u)
        vp[u] = fmaxf(fmaf(vp[u], sscale[c0+u], sscale[CH+c0+u]), 0.f);
      *(float4*)&atile[wave][rr*CH + c0] = v;
    }
    __syncthreads();
    v8f acc[4] = {};
    WMMA_TILE(acc, atile[wave], w2s, lh, hb)
#pragma unroll
    for (int nc = 0; nc < 4; ++nc) {
      const int nch = nc*16 + lh;
      const float bias = b2[nch];
      float mx = -3.4e38f;
#pragma unroll
      for (int r = 0; r < 8; ++r) mx = fmaxf(mx, acc[nc][r] + bias);
      mx = fmaxf(mx, __shfl_xor(mx, 16, 32));
      if (hb == 0) res[lp*CH + nch] = mx;
    }
    __syncthreads();
  }
  // coalesced transpose-out: out[(b*C + c)*N + n]
  {
    const int t = tid;                  // 256 threads, 4 floats each = 1024
    const int c = t >> 2;
    const int l0 = (t & 3) * 4;
    float4 o;
    float* op = (float*)&o;
#pragma unroll
    for (int u = 0; u < 4; ++u) op[u] = res[(l0 + u)*CH + c];
    *(float4*)(out + ((size_t)(b*CH + c))*NPT + nbase + l0) = o;
  }
}

// ---------------------------------------------------------------------------
extern "C" void kernel_launch(void* const* d_in, const int* in_sizes, int n_in,
                              void* d_out, int out_size, void* d_ws, size_t ws_size,
                              hipStream_t stream) {
  (void)in_sizes; (void)n_in; (void)out_size; (void)ws_size;
  const float* xyz   = (const float*)d_in[0];
  const float* W1    = (const float*)d_in[1];
  const float* b1    = (const float*)d_in[2];
  const float* W2    = (const float*)d_in[3];
  const float* b2    = (const float*)d_in[4];
  const float* gamma = (const float*)d_in[5];
  const float* beta  = (const float*)d_in[6];
  float* out = (float*)d_out;

  char* ws = (char*)d_ws;
  int*   knn_idx  = (int*)ws;                                   // 1 MB
  float* hbuf     = (float*)(ws + (size_t)(1 << 20));           // 64 MB
  float* partial1 = (float*)(ws + (size_t)(1 << 20) + (size_t)ROWS*CH*4);
  float* partial2 = partial1 + 1024*2*CH;                       // 512 KB after p1
  float* stats1   = partial2 + 2048*2*CH;
  float* stats2   = stats1 + 2*CH;

  const float inv_count = 1.0f / (float)ROWS;

  knn_kernel     <<<BSZ*(NPT/64), 256, 0, stream>>>(xyz, knn_idx);
  conv1_kernel   <<<ROWS/256,     256, 0, stream>>>(xyz, knn_idx, W1, b1, hbuf, partial1);
  bn_reduce_kernel<<<1, CH, 0, stream>>>(partial1, 1024, inv_count, gamma, beta, stats1);
  conv_mid_kernel<<<ROWS/128,     256, 0, stream>>>(W2, b2, stats1, hbuf, partial2);
  bn_reduce_kernel<<<1, CH, 0, stream>>>(partial2, 2048, inv_count, gamma, beta, stats2);
  conv_out_kernel<<<(BSZ*NPT)/16, 256, 0, stream>>>(W2, b2, stats2, hbuf, out);
}